// SolveScheduling_17738214933078
// MI455X (gfx1250) — compile-verified
//
#include <hip/hip_runtime.h>
#include <stdint.h>

// Problem constants (match reference)
#define GS_     5.0f
#define GE_     0.5f
#define C_RAMP_ 0.4f
#define N_ITERS_ 600
#define ETA_    0.25f
#define M_      128
#define N_      128
#define ROWPAD  132   // floats per padded LDS row: 528B, 16B aligned, bank-conflict-free columns

#if defined(__AMDGCN__) && __has_builtin(__builtin_amdgcn_global_load_async_to_lds_b128)
#define HAVE_ASYNC_LDS 1
#else
#define HAVE_ASYNC_LDS 0
#endif

#if HAVE_ASYNC_LDS
typedef int v4i_t __attribute__((vector_size(16)));
typedef __attribute__((address_space(1))) v4i_t* as1_v4i_ptr;
typedef __attribute__((address_space(3))) v4i_t* as3_v4i_ptr;
#endif

__device__ static inline void wait_async_zero() {
#if HAVE_ASYNC_LDS
#if __has_builtin(__builtin_amdgcn_s_wait_asynccnt)
    __builtin_amdgcn_s_wait_asynccnt(0);
#else
    asm volatile("s_wait_asynccnt 0" ::: "memory");
#endif
#endif
}

__global__ __launch_bounds__(128)
void SolveScheduling_kernel(const float* __restrict__ mu,
                            const float* __restrict__ sig,
                            const float* __restrict__ eps,
                            float* __restrict__ out) {
    // LDS: sorted sample block (row-major [s][i], padded rows) + exchange buffers
    __shared__ __attribute__((aligned(16))) float ys[M_ * ROWPAD];
    __shared__ float wbuf[N_];
    __shared__ float zbuf[N_ + 1];

    const int tid = threadIdx.x;     // coordinate i (and staging row s)
    const int b   = blockIdx.x;      // batch row
    const float slope = (GS_ + GE_) / (float)M_;

    if (tid == 0) zbuf[N_] = 0.0f;   // sentinel read by tid==127 (value unused)

    // ---------------- Phase 1: stage eps[b,:,:] (64KB) into LDS ----------------
    const float* eps_b = eps + (size_t)b * (M_ * N_);
#if HAVE_ASYNC_LDS
    // Thread tid copies global row s=tid (512B) as 32 async B128 transfers.
    for (int c = 0; c < 32; ++c) {
        const float* g = eps_b + tid * N_ + c * 4;
        float*       l = ys + tid * ROWPAD + c * 4;
        __builtin_amdgcn_global_load_async_to_lds_b128(
            (as1_v4i_ptr)(uintptr_t)g,
            (as3_v4i_ptr)(uint32_t)(uintptr_t)l,
            /*imm offset*/0, /*cpol*/0);
    }
    wait_async_zero();
#else
    for (int c = 0; c < 32; ++c) {
        const float4 v = *(const float4*)(eps_b + tid * N_ + c * 4);
        *(float4*)(ys + tid * ROWPAD + c * 4) = v;
    }
#endif
    __syncthreads();

    // ---------------- Phase 2: y = mu + sig*eps on own column; mean ----------------
    const float mui = mu[(size_t)b * N_ + tid];
    const float sgi = sig[(size_t)b * N_ + tid];
    float sum = 0.0f;
    for (int s = 0; s < M_; ++s) {
        float yv = fmaf(sgi, ys[s * ROWPAD + tid], mui);
        ys[s * ROWPAD + tid] = yv;
        sum += yv;
    }
    const float A0 = sum * (1.0f / (float)M_) + GS_;   // mean_y + GS

    // ---------------- Phase 3: per-thread bitonic sort of own column (ascending) ----
    for (int k = 2; k <= M_; k <<= 1) {
        for (int j = k >> 1; j > 0; j >>= 1) {
#pragma clang loop unroll_count(4)
            for (int s = 0; s < M_; ++s) {
                int l = s ^ j;
                if (l > s) {
                    float a = ys[s * ROWPAD + tid];
                    float c2 = ys[l * ROWPAD + tid];
                    bool up = ((s & k) == 0);
                    bool sw = up ? (a > c2) : (a < c2);
                    if (sw) {
                        ys[s * ROWPAD + tid] = c2;
                        ys[l * ROWPAD + tid] = a;
                    }
                }
            }
        }
    }
    // Own column only -> no barrier needed before FISTA (first barrier covers wbuf)

    // Exact separable prox: argmin_z sum g_i(z_i) + v_i z_i via breakpoint search
    auto zstar = [&](float v) -> float {
        float q = A0 - v;
        int lo = 0, hi = M_;
        // k = searchsorted(t, q, side='right'), t[m] = ys[m] + slope*(m+1)
        while (lo < hi) {
            int mid = (lo + hi) >> 1;
            float tm = fmaf(slope, (float)(mid + 1), ys[mid * ROWPAD + tid]);
            if (tm <= q) lo = mid + 1; else hi = mid;
        }
        float zseg = fmaf(-slope, (float)lo, q);
        float hiv  = (lo < M_) ? ys[lo * ROWPAD + tid] : __builtin_inff();
        return fminf(zseg, hiv);
    };

    // ---------------- Phase 4: accelerated projected dual ascent (FISTA) -----------
    float lp = 0.0f, lm = 0.0f, yp = 0.0f, ym = 0.0f, tk = 1.0f;
    const bool edge = (tid < N_ - 1);

    for (int it = 0; it < N_ITERS_; ++it) {
        float w = edge ? (yp - ym) : 0.0f;     // dual edge variable
        wbuf[tid] = w;
        __syncthreads();
        float wprev = (tid > 0) ? wbuf[tid - 1] : 0.0f;
        float v = wprev - w;                   // v = Dt(yp - ym)

        float z = zstar(v);
        zbuf[tid] = z;
        __syncthreads();
        float dz = zbuf[tid + 1] - z;          // D(z); tid==127 reads sentinel (unused)

        float lpn = fmaxf(fmaf(ETA_,  dz - C_RAMP_, yp), 0.0f);
        float lmn = fmaxf(fmaf(ETA_, -dz - C_RAMP_, ym), 0.0f);
        float tk1 = 0.5f * (1.0f + sqrtf(fmaf(4.0f * tk, tk, 1.0f)));
        float beta = (tk - 1.0f) / tk1;
        yp = fmaf(beta, lpn - lp, lpn);
        ym = fmaf(beta, lmn - lm, lmn);
        lp = lpn; lm = lmn; tk = tk1;
    }

    // ---------------- Final: z = zstar(Dt(lp - lm)) --------------------------------
    {
        float w = edge ? (lp - lm) : 0.0f;
        wbuf[tid] = w;
        __syncthreads();
        float wprev = (tid > 0) ? wbuf[tid - 1] : 0.0f;
        out[(size_t)b * N_ + tid] = zstar(wprev - w);
    }
}

extern "C" void kernel_launch(void* const* d_in, const int* in_sizes, int n_in,
                              void* d_out, int out_size, void* d_ws, size_t ws_size,
                              hipStream_t stream) {
    const float* mu  = (const float*)d_in[0];   // (B, n)
    const float* sig = (const float*)d_in[1];   // (B, n)
    const float* eps = (const float*)d_in[2];   // (B, M, n)
    float* out = (float*)d_out;                 // (B, n)
    const int B = in_sizes[0] / N_;             // 128
    SolveScheduling_kernel<<<dim3(B), dim3(N_), 0, stream>>>(mu, sig, eps, out);
}